// CRF_34643206210294
// MI455X (gfx1250) — compile-verified
//
#include <hip/hip_runtime.h>
#include <hip/hip_bf16.h>
#include <stdint.h>

// ---------------------------------------------------------------------------
// CRF loss on gfx1250:  out[b] = forward_logZ[b] - gold_score[b]
// Forward scan reformulated as a rescaled GEMM per step so the T x T
// logsumexp runs on V_WMMA_F32_16X16X4_F32 (full fp32 matrix pipe).
// Critical-path features: loop-invariant B operand (exp(trans)) held in
// registers across the whole scan, dual WMMA accumulators, shuffle+ds_max
// row-max fused into the epilogue, register-fed pT rescale with packed
// ds_store_b128, double-buffered feats slab loaded in the WMMA shadow,
// 2 barriers/step.
// ---------------------------------------------------------------------------

#define CRF_B 256
#define CRF_S 1024
#define CRF_T 73
#define KP 76   // K (tags-in) padded to multiple of 4
#define CP 80   // column pad for LDS rows
#define NEG_INF (-3.4e38f)

typedef __attribute__((ext_vector_type(2))) float v2f;
typedef __attribute__((ext_vector_type(8))) float v8f;

// ----------------------------- gold score ---------------------------------
__global__ __launch_bounds__(256) void crf_gold_kernel(
    const float* __restrict__ feats, const unsigned char* __restrict__ mask,
    const int* __restrict__ tags, const float* __restrict__ cdt,
    const float* __restrict__ start_t, const float* __restrict__ stop_t,
    const int* __restrict__ types0, const int* __restrict__ types1,
    float* __restrict__ gold) {
  __shared__ float trans_sh[CRF_T * CRF_T];
  __shared__ float red_s[256];
  __shared__ int red_len[256];

  const int b = blockIdx.x;
  const int tid = threadIdx.x;

  for (int idx = tid; idx < CRF_T * CRF_T; idx += 256)
    trans_sh[idx] = cdt[types0[idx] * 5 + types1[idx]];
  __syncthreads();

  const float* fb = feats + (size_t)b * CRF_S * CRF_T;
  const int* tb = tags + (size_t)b * CRF_S;
  const unsigned char* mb = mask + (size_t)b * CRF_S;

  float s = 0.f;
  int len = 0;
  for (int t = tid; t < CRF_S; t += 256) {
    const int on = mb[t] ? 1 : 0;
    const float m = (float)on;
    len += on;
    const int tg = tb[t];
    s += fb[(size_t)t * CRF_T + tg] * m;
    if (t >= 1) s += trans_sh[tb[t - 1] * CRF_T + tg] * m;
  }
  red_s[tid] = s;
  red_len[tid] = len;
  __syncthreads();
  for (int off = 128; off > 0; off >>= 1) {
    if (tid < off) {
      red_s[tid] += red_s[tid + off];
      red_len[tid] += red_len[tid + off];
    }
    __syncthreads();
  }
  if (tid == 0) {
    int L = red_len[0];
    if (L < 1) L = 1;
    gold[b] = red_s[0] + start_t[tb[0]] + stop_t[tb[L - 1]];
  }
}

// --------------------- forward algorithm (WMMA scan) -----------------------
// grid = B/16 blocks, block = 160 threads = 5 wave32 (one wave per 16-wide
// N-tile of the 73 output tags).
__global__ __launch_bounds__(160) void crf_forward_kernel(
    const float* __restrict__ feats, const unsigned char* __restrict__ mask,
    const float* __restrict__ cdt, const float* __restrict__ start_t,
    const float* __restrict__ stop_t, const int* __restrict__ types0,
    const int* __restrict__ types1, const float* __restrict__ gold,
    float* __restrict__ out) {
  __shared__ float E_sh[KP][CP];                    // exp(trans), zero padded
  __shared__ __align__(16) float pT_sh[KP][16];     // exp(part-Pm), transposed
  __shared__ float part_sh[16][CP];                 // running log partition
  __shared__ float f_sh[2][16][CP];                 // double-buffered feats
  __shared__ float Pm_sh[2][16];                    // double-buffered row max
  __shared__ unsigned char m_sh[2][16];

  const int tid = threadIdx.x;
  const int b0 = blockIdx.x * 16;

  // ---- one-time: E = exp(trans), padded with zeros ----
  for (int idx = tid; idx < KP * CP; idx += 160) {
    const int k = idx / CP, j = idx % CP;
    float e = 0.f;
    if (k < CRF_T && j < CRF_T)
      e = __expf(cdt[types0[k * CRF_T + j] * 5 + types1[k * CRF_T + j]]);
    E_sh[k][j] = e;
  }
  // ---- init partition: feats[:,0,:] + start ----
  for (int idx = tid; idx < 16 * CRF_T; idx += 160) {
    const int r = idx / CRF_T, c = idx % CRF_T;
    part_sh[r][c] = feats[((size_t)(b0 + r) * CRF_S) * CRF_T + c] + start_t[c];
  }
  // ---- preload feats/mask slab for t=1 (buffer 1) ----
  for (int idx = tid; idx < 16 * CRF_T; idx += 160) {
    const int r = idx / CRF_T, c = idx % CRF_T;
    f_sh[1][r][c] = feats[((size_t)(b0 + r) * CRF_S + 1) * CRF_T + c];
  }
  if (tid < 16) {
    m_sh[1][tid] = mask[(size_t)(b0 + tid) * CRF_S + 1];
    Pm_sh[1][tid] = NEG_INF;  // ds_max target for step t=1
  }
  __syncthreads();
  if (tid < 16) {  // one-time serial row max is fine
    float mx = NEG_INF;
    for (int j = 0; j < CRF_T; ++j) mx = fmaxf(mx, part_sh[tid][j]);
    Pm_sh[0][tid] = mx;
  }
  __syncthreads();
  // initial pT fill; rows k>=73 stay zero forever (never rewritten in-loop)
  for (int idx = tid; idx < KP * 16; idx += 160) {
    const int k = idx >> 4, r = idx & 15;
    pT_sh[k][r] = (k < CRF_T) ? __expf(part_sh[r][k] - Pm_sh[0][r]) : 0.f;
  }
  __syncthreads();

  const int lane = tid & 31;
  const int lane15 = lane & 15;
  const int jb = (tid >> 5) * 16;        // wave's N-tile base (0..64)
  const int nL = jb + lane15;            // this lane's output column
  const int mbase = (lane < 16) ? 0 : 8; // C/D: VGPR r -> M=r (+8 upper half)
  const int ksel = (lane < 16) ? 0 : 2;  // A/B: upper half holds K+2/K+3
  const bool incol = (nL < CRF_T);

  // ---- hoist the loop-invariant B operand (E tile) into registers ----
  v2f breg[19];
#pragma unroll
  for (int i = 0; i < 19; ++i) {
    const int ka = i * 4 + ksel;
    breg[i].x = E_sh[ka][nL];
    breg[i].y = E_sh[ka + 1][nL];
  }

  for (int t = 1; t < CRF_S; ++t) {
    const int bt = t & 1;        // current slab / Pm_new buffer
    const int bn = bt ^ 1;       // next slab / Pm_used buffer

    // ---- phase 1: WMMA (dual accumulators, register B operand) ----
    v8f acc0 = {}, acc1 = {};
#pragma unroll
    for (int i = 0; i < 18; i += 2) {
      {
        const int ka = i * 4 + ksel;
        v2f a;
        a.x = pT_sh[ka][lane15];
        a.y = pT_sh[ka + 1][lane15];
        acc0 = __builtin_amdgcn_wmma_f32_16x16x4_f32(false, a, false, breg[i],
                                                     (short)0, acc0, false,
                                                     false);
      }
      {
        const int ka = (i + 1) * 4 + ksel;
        v2f a;
        a.x = pT_sh[ka][lane15];
        a.y = pT_sh[ka + 1][lane15];
        acc1 = __builtin_amdgcn_wmma_f32_16x16x4_f32(
            false, a, false, breg[i + 1], (short)0, acc1, false, false);
      }
    }
    {  // tail k-step 18 (k 72..75, zero padded)
      const int ka = 72 + ksel;
      v2f a;
      a.x = pT_sh[ka][lane15];
      a.y = pT_sh[ka + 1][lane15];
      acc0 = __builtin_amdgcn_wmma_f32_16x16x4_f32(false, a, false, breg[18],
                                                   (short)0, acc0, false,
                                                   false);
    }

    // load feats/mask for t+1 into the other buffer (hides under WMMA)
    if (t + 1 < CRF_S) {
      for (int idx = tid; idx < 16 * CRF_T; idx += 160) {
        const int r = idx / CRF_T, c = idx % CRF_T;
        f_sh[bn][r][c] = feats[((size_t)(b0 + r) * CRF_S + t + 1) * CRF_T + c];
      }
      if (tid < 16) {
        m_sh[bn][tid] = mask[(size_t)(b0 + tid) * CRF_S + t + 1];
        if (t + 2 < CRF_S)
          __builtin_prefetch(
              &feats[((size_t)(b0 + tid) * CRF_S + t + 2) * CRF_T], 0, 0);
      }
    }

    // ---- epilogue: part' = f + Pm + log(V); fused row-max reduction ----
    float vals[8];  // kept for the register-fed pT rescale after the barrier
    float red[8];
#pragma unroll
    for (int r = 0; r < 8; ++r) {
      const int row = mbase + r;
      const float vsum = acc0[r] + acc1[r];
      const float oldv = part_sh[row][nL];
      const float np = f_sh[bt][row][nL] + Pm_sh[bn][row] + __logf(vsum);
      const float vsel = m_sh[bt][row] ? np : oldv;
      if (incol) part_sh[row][nL] = vsel;
      vals[r] = vsel;
      red[r] = incol ? vsel : NEG_INF;
    }
    // half-wave (16-lane) max reduce; offsets <16 never cross halves
#pragma unroll
    for (int off = 1; off < 16; off <<= 1) {
#pragma unroll
      for (int r = 0; r < 8; ++r)
        red[r] = fmaxf(red[r], __shfl_xor(red[r], off, 32));
    }
    if (lane15 == 0) {  // lanes 0 and 16: one ds_max per row per wave
#pragma unroll
      for (int r = 0; r < 8; ++r)
        atomicMax(&Pm_sh[bt][mbase + r], red[r]);
    }
    __syncthreads();

    // ---- phase 2: register-fed pT rescale (2 x ds_store_b128 per lane) ----
    if (incol) {
      float4 lo, hi;
      lo.x = __expf(vals[0] - Pm_sh[bt][mbase + 0]);
      lo.y = __expf(vals[1] - Pm_sh[bt][mbase + 1]);
      lo.z = __expf(vals[2] - Pm_sh[bt][mbase + 2]);
      lo.w = __expf(vals[3] - Pm_sh[bt][mbase + 3]);
      hi.x = __expf(vals[4] - Pm_sh[bt][mbase + 4]);
      hi.y = __expf(vals[5] - Pm_sh[bt][mbase + 5]);
      hi.z = __expf(vals[6] - Pm_sh[bt][mbase + 6]);
      hi.w = __expf(vals[7] - Pm_sh[bt][mbase + 7]);
      float4* dst = (float4*)&pT_sh[nL][mbase];
      dst[0] = lo;
      dst[1] = hi;
    }
    if (tid < 16) Pm_sh[bn][tid] = NEG_INF;  // re-arm for step t+1
    __syncthreads();
  }

  // ---- finalize: logsumexp(part + stop) - gold ----
  if (tid < 16) {
    float mx = NEG_INF;
    for (int j = 0; j < CRF_T; ++j)
      mx = fmaxf(mx, part_sh[tid][j] + stop_t[j]);
    float sm = 0.f;
    for (int j = 0; j < CRF_T; ++j)
      sm += __expf(part_sh[tid][j] + stop_t[j] - mx);
    out[b0 + tid] = mx + __logf(sm) - gold[b0 + tid];
  }
}

// ---------------------------------------------------------------------------
extern "C" void kernel_launch(void* const* d_in, const int* in_sizes, int n_in,
                              void* d_out, int out_size, void* d_ws,
                              size_t ws_size, hipStream_t stream) {
  const float* feats = (const float*)d_in[0];
  const unsigned char* mask = (const unsigned char*)d_in[1];  // jnp bool = u8
  const int* tags = (const int*)d_in[2];
  const float* cdt = (const float*)d_in[3];
  const float* start_t = (const float*)d_in[4];
  const float* stop_t = (const float*)d_in[5];
  const int* types0 = (const int*)d_in[6];
  const int* types1 = (const int*)d_in[7];

  float* gold = (float*)d_ws;  // [B] scratch
  float* out = (float*)d_out;  // [B]

  crf_gold_kernel<<<CRF_B, 256, 0, stream>>>(feats, mask, tags, cdt, start_t,
                                             stop_t, types0, types1, gold);
  crf_forward_kernel<<<CRF_B / 16, 160, 0, stream>>>(
      feats, mask, cdt, start_t, stop_t, types0, types1, gold, out);
}